// DilatedMDTA_75316546503098
// MI455X (gfx1250) — compile-verified
//
#include <hip/hip_runtime.h>
#include <hip/hip_bf16.h>

#define DIM  512
#define NPIX 4096
#define HEADS 8
#define CPH  64

typedef __attribute__((ext_vector_type(2))) float v2f;
typedef __attribute__((ext_vector_type(8))) float v8f;

__device__ __forceinline__ v8f wmma_f32(v2f a, v2f b, v8f c) {
  // D = A(16x4 f32) * B(4x16 f32) + C(16x16 f32)
  // 8-arg form: (neg_a, A, neg_b, B, c_mod, C, reuse_a, reuse_b)
  return __builtin_amdgcn_wmma_f32_16x16x4_f32(false, a, false, b, (short)0, c,
                                               false, false);
}

// ---------------------------------------------------------------------------
// Pointwise 1x1 conv as GEMM: Y[b,o,n] = sum_c W[o,c] * X[b,c,n]
// grid (NPIX/128, DIM/16, B), block 256 = 8 waves, one 16x16 tile per wave.
// ---------------------------------------------------------------------------
__global__ __launch_bounds__(256) void pw_gemm_kernel(
    const float* __restrict__ X, const float* __restrict__ W,
    float* __restrict__ Y)
{
  const int lane = threadIdx.x & 31;
  const int wave = threadIdx.x >> 5;
  const int row  = lane & 15;   // M (A) / N (B,C) index within tile
  const int grp  = lane >> 4;   // selects K pair (A/B), M halves (C/D)
  const int n0   = blockIdx.x * 128 + wave * 16;
  const int o0   = blockIdx.y * 16;
  const size_t bOff = (size_t)blockIdx.z * DIM * NPIX;
  const float* Xb = X + bOff;

  v8f acc = {0.f, 0.f, 0.f, 0.f, 0.f, 0.f, 0.f, 0.f};
#pragma unroll 4
  for (int k = 0; k < DIM; k += 4) {
    // A tile: W[o0+row][k+2g .. k+2g+1]  (contiguous float2)
    v2f a = *(const v2f*)(W + (size_t)(o0 + row) * DIM + k + 2 * grp);
    // B tile: X[b][k+2g + j][n0+row], j=0,1 (stride NPIX)
    const float* xp = Xb + (size_t)(k + 2 * grp) * NPIX + n0 + row;
    v2f b; b.x = xp[0]; b.y = xp[NPIX];
    acc = wmma_f32(a, b, acc);
  }
  float* Yb = Y + bOff;
#pragma unroll
  for (int r = 0; r < 8; ++r)
    Yb[(size_t)(o0 + r + 8 * grp) * NPIX + n0 + row] = acc[r];
}

// Same GEMM + residual add (final projection), writes d_out.
__global__ __launch_bounds__(256) void pw_gemm_res_kernel(
    const float* __restrict__ X, const float* __restrict__ W,
    const float* __restrict__ Res, float* __restrict__ Y)
{
  const int lane = threadIdx.x & 31;
  const int wave = threadIdx.x >> 5;
  const int row  = lane & 15;
  const int grp  = lane >> 4;
  const int n0   = blockIdx.x * 128 + wave * 16;
  const int o0   = blockIdx.y * 16;
  const size_t bOff = (size_t)blockIdx.z * DIM * NPIX;
  const float* Xb = X + bOff;

  v8f acc = {0.f, 0.f, 0.f, 0.f, 0.f, 0.f, 0.f, 0.f};
#pragma unroll 4
  for (int k = 0; k < DIM; k += 4) {
    v2f a = *(const v2f*)(W + (size_t)(o0 + row) * DIM + k + 2 * grp);
    const float* xp = Xb + (size_t)(k + 2 * grp) * NPIX + n0 + row;
    v2f b; b.x = xp[0]; b.y = xp[NPIX];
    acc = wmma_f32(a, b, acc);
  }
  const float* Rb = Res + bOff;
  float* Yb = Y + bOff;
#pragma unroll
  for (int r = 0; r < 8; ++r) {
    size_t off = (size_t)(o0 + r + 8 * grp) * NPIX + n0 + row;
    Yb[off] = acc[r] + Rb[off];
  }
}

// ---------------------------------------------------------------------------
// Depthwise 3x3, dilation 2, ReflectionPad(2): taps at h-2,h,h+2 (reflected).
// ---------------------------------------------------------------------------
__global__ __launch_bounds__(256) void dwconv_kernel(
    const float* __restrict__ Yin, const float* __restrict__ WD,
    float* __restrict__ Yout)
{
  size_t idx = (size_t)blockIdx.x * 256 + threadIdx.x;   // over B*DIM*NPIX
  int pos = (int)(idx & (NPIX - 1));
  int c   = (int)((idx >> 12) & (DIM - 1));
  int h = pos >> 6, w = pos & 63;
  const float* src = Yin + (idx & ~(size_t)(NPIX - 1));  // channel plane base

  float wgt[9];
#pragma unroll
  for (int t = 0; t < 9; ++t) wgt[t] = WD[c * 9 + t];

  float sum = 0.f;
#pragma unroll
  for (int i = 0; i < 3; ++i) {
    int hh = h + (i - 1) * 2;
    hh = hh < 0 ? -hh : (hh > 63 ? 126 - hh : hh);
#pragma unroll
    for (int j = 0; j < 3; ++j) {
      int ww = w + (j - 1) * 2;
      ww = ww < 0 ? -ww : (ww > 63 ? 126 - ww : ww);
      sum = fmaf(wgt[i * 3 + j], src[hh * 64 + ww], sum);
    }
  }
  Yout[idx] = sum;
}

// ---------------------------------------------------------------------------
// Channel attention per (b,head): E = Q K^T * temp, softmax rows, O = A V.
// One 512-thread block (16 waves) per (b,head). Q/K/V: [bh][64][4096].
// ---------------------------------------------------------------------------
__global__ __launch_bounds__(512) void attn_kernel(
    const float* __restrict__ Q, const float* __restrict__ K,
    const float* __restrict__ V, const float* __restrict__ temp,
    float* __restrict__ O)
{
  __shared__ float e[64][65];   // 65 stride: avoid LDS bank conflicts
  const int bh   = blockIdx.x;
  const int head = bh & (HEADS - 1);
  const int lane = threadIdx.x & 31;
  const int wave = threadIdx.x >> 5;
  const int row  = lane & 15;
  const int grp  = lane >> 4;
  const size_t base = (size_t)bh * CPH * NPIX;
  const float* qp = Q + base;
  const float* kp = K + base;
  const float* vp = V + base;

  // ---- Phase 1: energy tiles (16 waves, one 16x16 tile each, K=4096) ----
  {
    const int ci = wave >> 2, di = wave & 3;
    const float* qrow = qp + (size_t)(ci * 16 + row) * NPIX;
    const float* krow = kp + (size_t)(di * 16 + row) * NPIX;
    v8f acc = {0.f, 0.f, 0.f, 0.f, 0.f, 0.f, 0.f, 0.f};
#pragma unroll 4
    for (int k = 0; k < NPIX; k += 4) {
      v2f a = *(const v2f*)(qrow + k + 2 * grp);      // Q rows (M)
      v2f b = *(const v2f*)(krow + k + 2 * grp);      // K rows = B cols (N)
      acc = wmma_f32(a, b, acc);
    }
    float tscale = temp[head];
#pragma unroll
    for (int r = 0; r < 8; ++r)
      e[ci * 16 + r + 8 * grp][di * 16 + row] = acc[r] * tscale;
  }
  __syncthreads();

  // ---- Softmax over rows of the 64x64 energy matrix ----
  if (threadIdx.x < 64) {
    const int r = threadIdx.x;
    float m = -3.4e38f;
#pragma unroll 8
    for (int j = 0; j < 64; ++j) m = fmaxf(m, e[r][j]);
    float s = 0.f;
#pragma unroll 8
    for (int j = 0; j < 64; ++j) { float v = __expf(e[r][j] - m); e[r][j] = v; s += v; }
    float inv = 1.f / s;
#pragma unroll 8
    for (int j = 0; j < 64; ++j) e[r][j] *= inv;
  }
  __syncthreads();

  // ---- Phase 2: O = attn(LDS) * V, 1024 tiles round-robined over waves ----
  float* op = O + base;
  for (int tile = wave; tile < 1024; tile += 16) {
    const int ci = tile & 3;
    const int n0 = (tile >> 2) * 16;
    v8f acc = {0.f, 0.f, 0.f, 0.f, 0.f, 0.f, 0.f, 0.f};
#pragma unroll
    for (int d = 0; d < 64; d += 4) {
      v2f a; a.x = e[ci * 16 + row][d + 2 * grp];
             a.y = e[ci * 16 + row][d + 2 * grp + 1];
      const float* vptr = vp + (size_t)(d + 2 * grp) * NPIX + n0 + row;
      v2f b; b.x = vptr[0]; b.y = vptr[NPIX];
      acc = wmma_f32(a, b, acc);
    }
#pragma unroll
    for (int r = 0; r < 8; ++r)
      op[(size_t)(ci * 16 + r + 8 * grp) * NPIX + n0 + row] = acc[r];
  }
}

// ---------------------------------------------------------------------------
extern "C" void kernel_launch(void* const* d_in, const int* in_sizes, int n_in,
                              void* d_out, int out_size, void* d_ws, size_t ws_size,
                              hipStream_t stream)
{
  const float* x    = (const float*)d_in[0];   // [16,512,64,64]
  const float* fw1  = (const float*)d_in[1];
  const float* fwd_ = (const float*)d_in[2];
  const float* gw1  = (const float*)d_in[3];
  const float* gwd  = (const float*)d_in[4];
  const float* hw1  = (const float*)d_in[5];
  const float* hwd  = (const float*)d_in[6];
  const float* ow   = (const float*)d_in[7];
  const float* temp = (const float*)d_in[8];

  const size_t nElem = (size_t)16 * DIM * NPIX;
  float* tmp = (float*)d_ws;          // branch temp, later attention output
  float* qb  = tmp + nElem;
  float* kb  = qb + nElem;
  float* vb  = kb + nElem;

  dim3 gG(NPIX / 128, DIM / 16, 16), bG(256);
  dim3 gD((unsigned)(nElem / 256)),  bD(256);

  // f branch -> q
  pw_gemm_kernel<<<gG, bG, 0, stream>>>(x, fw1, tmp);
  dwconv_kernel <<<gD, bD, 0, stream>>>(tmp, fwd_, qb);
  // g branch -> k
  pw_gemm_kernel<<<gG, bG, 0, stream>>>(x, gw1, tmp);
  dwconv_kernel <<<gD, bD, 0, stream>>>(tmp, gwd, kb);
  // h branch -> v
  pw_gemm_kernel<<<gG, bG, 0, stream>>>(x, hw1, tmp);
  dwconv_kernel <<<gD, bD, 0, stream>>>(tmp, hwd, vb);
  // attention (writes into tmp, which is free now)
  attn_kernel<<<dim3(16 * HEADS), dim3(512), 0, stream>>>(qb, kb, vb, temp, tmp);
  // final projection + residual
  pw_gemm_res_kernel<<<gG, bG, 0, stream>>>(tmp, ow, x, (float*)d_out);
}